// GCN_Encoder_32727650795996
// MI455X (gfx1250) — compile-verified
//
#include <hip/hip_runtime.h>

typedef __attribute__((ext_vector_type(2))) float v2f;
typedef __attribute__((ext_vector_type(8))) float v8f;

#define F_IN  128
#define F_HID 16

// ---------------- degree / normalization ----------------
__global__ void deg_kernel(const long long* __restrict__ ei, float* __restrict__ deg, int E) {
    int e = blockIdx.x * blockDim.x + threadIdx.x;
    if (e >= E) return;
    int dst = (int)ei[(size_t)E + e];
    unsafeAtomicAdd(&deg[dst], 1.0f);
}

__global__ void dinv_kernel(float* __restrict__ deg, int N) {
    int i = blockIdx.x * blockDim.x + threadIdx.x;
    if (i >= N) return;
    deg[i] = rsqrtf(deg[i] + 1.0f);   // +1 = self loop; always > 0
}

// ---------------- GEMM1: H1 = X (N x 128) * W1 (128 x 16) ----------------
// One wave computes one 16x16 output tile (16 rows, all 16 hidden features).
__global__ __launch_bounds__(256) void gemm1_kernel(
        const float* __restrict__ x, const float* __restrict__ W1,
        float* __restrict__ h1, int N) {
    int wave = threadIdx.x >> 5;
    int lane = threadIdx.x & 31;
    int tile = blockIdx.x * (blockDim.x >> 5) + wave;
    if (tile >= (N >> 4)) return;          // wave-uniform: EXEC stays all-ones
    int row0 = tile << 4;
    int half = lane >> 4;                  // 0: K pair {0,1}, 1: K pair {2,3}
    int m    = lane & 15;                  // A row / B,D col

    const float* arow = x + (size_t)(row0 + m) * F_IN;
    v8f c = {};
    #pragma unroll 8
    for (int k0 = 0; k0 < F_IN; k0 += 4) {
        v2f a, b;
        a.x = arow[k0 + 2 * half];
        a.y = arow[k0 + 2 * half + 1];
        b.x = W1[(size_t)(k0 + 2 * half) * F_HID + m];
        b.y = W1[(size_t)(k0 + 2 * half + 1) * F_HID + m];
        c = __builtin_amdgcn_wmma_f32_16x16x4_f32(false, a, false, b,
                                                  (short)0, c, false, false);
    }
    // D layout: VGPR r -> row (r + 8*half), col = m
    #pragma unroll
    for (int r = 0; r < 8; ++r)
        h1[(size_t)(row0 + r + 8 * half) * F_HID + m] = c[r];
}

// ---------------- edge aggregation in 16-dim space ----------------
// 4 threads per edge; each handles 4 features via a b128 gather + 4 atomics.
// agg[dst] += dinv[src]*dinv[dst]*h[src]
__global__ void agg_kernel(const long long* __restrict__ ei,
                           const float* __restrict__ dinv,
                           const float* __restrict__ h,
                           float* __restrict__ agg, int E) {
    int t = blockIdx.x * blockDim.x + threadIdx.x;   // up to E*4 = 6.4M
    if (t >= E * 4) return;
    int e  = t >> 2;
    int q  = (t & 3) << 2;                            // feature group offset 0/4/8/12
    int src = (int)ei[e];
    int dst = (int)ei[(size_t)E + e];
    float norm = dinv[src] * dinv[dst];
    const float4 v = *(const float4*)(h + (size_t)src * F_HID + q);
    float* ap = agg + (size_t)dst * F_HID + q;
    unsafeAtomicAdd(ap + 0, norm * v.x);
    unsafeAtomicAdd(ap + 1, norm * v.y);
    unsafeAtomicAdd(ap + 2, norm * v.z);
    unsafeAtomicAdd(ap + 3, norm * v.w);
}

// Hrelu = relu(agg1 + dinv^2 * h1 + b1)   (in place over agg1), float4-wide
__global__ void finalize1_kernel(float* __restrict__ agg1,
                                 const float* __restrict__ h1,
                                 const float* __restrict__ dinv,
                                 const float* __restrict__ b1, int NQ) {
    int t = blockIdx.x * blockDim.x + threadIdx.x;   // N*4 quads
    if (t >= NQ) return;
    int i = t >> 2;                                   // node
    int q = (t & 3) << 2;                             // feature group
    float d = dinv[i];
    float d2 = d * d;
    float4 a = *(float4*)(agg1 + (size_t)i * F_HID + q);
    float4 hv = *(const float4*)(h1 + (size_t)i * F_HID + q);
    float4 bv = *(const float4*)(b1 + q);
    a.x = fmaxf(a.x + d2 * hv.x + bv.x, 0.0f);
    a.y = fmaxf(a.y + d2 * hv.y + bv.y, 0.0f);
    a.z = fmaxf(a.z + d2 * hv.z + bv.z, 0.0f);
    a.w = fmaxf(a.w + d2 * hv.w + bv.w, 0.0f);
    *(float4*)(agg1 + (size_t)i * F_HID + q) = a;
}

// agg2 += dinv^2 * hrelu   (self-loop term of layer 2), float4-wide
__global__ void finalize2_kernel(float* __restrict__ agg2,
                                 const float* __restrict__ hrelu,
                                 const float* __restrict__ dinv, int NQ) {
    int t = blockIdx.x * blockDim.x + threadIdx.x;
    if (t >= NQ) return;
    int i = t >> 2;
    int q = (t & 3) << 2;
    float d = dinv[i];
    float d2 = d * d;
    float4 a  = *(float4*)(agg2 + (size_t)i * F_HID + q);
    float4 hv = *(const float4*)(hrelu + (size_t)i * F_HID + q);
    a.x += d2 * hv.x; a.y += d2 * hv.y; a.z += d2 * hv.z; a.w += d2 * hv.w;
    *(float4*)(agg2 + (size_t)i * F_HID + q) = a;
}

// ---------------- GEMM2: out = agg2 (N x 16) * W2 (16 x 128) + b2 ----------------
__global__ __launch_bounds__(256) void gemm2_kernel(
        const float* __restrict__ agg2, const float* __restrict__ W2,
        const float* __restrict__ b2, float* __restrict__ out, int N) {
    int wave = threadIdx.x >> 5;
    int lane = threadIdx.x & 31;
    int tile = blockIdx.x * (blockDim.x >> 5) + wave;
    if (tile >= (N >> 4)) return;          // wave-uniform
    int row0 = tile << 4;
    int half = lane >> 4;
    int m    = lane & 15;

    // A fragments for the whole K=16 reduction, reused across 8 N-tiles
    const float* arow = agg2 + (size_t)(row0 + m) * F_HID;
    v2f a[4];
    #pragma unroll
    for (int kk = 0; kk < 4; ++kk) {
        a[kk].x = arow[4 * kk + 2 * half];
        a[kk].y = arow[4 * kk + 2 * half + 1];
    }

    #pragma unroll
    for (int n0 = 0; n0 < F_IN; n0 += 16) {
        v8f c = {};
        #pragma unroll
        for (int kk = 0; kk < 4; ++kk) {
            int k = 4 * kk + 2 * half;
            v2f b;
            b.x = W2[(size_t)k * F_IN + n0 + m];
            b.y = W2[(size_t)(k + 1) * F_IN + n0 + m];
            c = __builtin_amdgcn_wmma_f32_16x16x4_f32(false, a[kk], false, b,
                                                      (short)0, c, false, false);
        }
        float bias = b2[n0 + m];
        #pragma unroll
        for (int r = 0; r < 8; ++r)
            out[(size_t)(row0 + r + 8 * half) * F_IN + n0 + m] = c[r] + bias;
    }
}

extern "C" void kernel_launch(void* const* d_in, const int* in_sizes, int n_in,
                              void* d_out, int out_size, void* d_ws, size_t ws_size,
                              hipStream_t stream) {
    const float*     x  = (const float*)d_in[0];
    const long long* ei = (const long long*)d_in[1];   // int64 [2, E]
    const float*     W1 = (const float*)d_in[2];
    const float*     b1 = (const float*)d_in[3];
    const float*     W2 = (const float*)d_in[4];
    const float*     b2 = (const float*)d_in[5];
    float*           out = (float*)d_out;

    const int N = in_sizes[0] / F_IN;   // 100000 (multiple of 16)
    const int E = in_sizes[1] / 2;      // 1600000

    // workspace layout (floats), 256-element aligned chunks
    float* ws = (float*)d_ws;
    size_t o = 0;
    auto take = [&](size_t n) { float* p = ws + o; o += (n + 255) & ~(size_t)255; return p; };
    float* dinv = take((size_t)N);              // degree, then dinv in place
    float* h1   = take((size_t)N * F_HID);      // X*W1
    float* agg1 = take((size_t)N * F_HID);      // layer-1 aggregate -> Hrelu (in place)
    float* agg2 = take((size_t)N * F_HID);      // layer-2 aggregate (16-dim)

    hipMemsetAsync(dinv, 0, (size_t)N * sizeof(float), stream);
    hipMemsetAsync(agg1, 0, (size_t)N * F_HID * sizeof(float), stream);
    hipMemsetAsync(agg2, 0, (size_t)N * F_HID * sizeof(float), stream);

    const int B = 256;
    const int NQ = N * 4;                           // node feature quads
    const int EQ = E * 4;                           // edge feature quads
    const int gemm_blocks = ((N >> 4) + 7) / 8;     // 8 waves (tiles) per block

    deg_kernel      <<<(E + B - 1) / B, B, 0, stream>>>(ei, dinv, E);
    dinv_kernel     <<<(N + B - 1) / B, B, 0, stream>>>(dinv, N);
    gemm1_kernel    <<<gemm_blocks, B, 0, stream>>>(x, W1, h1, N);
    agg_kernel      <<<(EQ + B - 1) / B, B, 0, stream>>>(ei, dinv, h1, agg1, E);
    finalize1_kernel<<<(NQ + B - 1) / B, B, 0, stream>>>(agg1, h1, dinv, b1, NQ);
    agg_kernel      <<<(EQ + B - 1) / B, B, 0, stream>>>(ei, dinv, agg1, agg2, E);
    finalize2_kernel<<<(NQ + B - 1) / B, B, 0, stream>>>(agg2, agg1, dinv, NQ);
    gemm2_kernel    <<<gemm_blocks, B, 0, stream>>>(agg2, W2, b2, out, N);
}